// VQ_layer_16612933500990
// MI455X (gfx1250) — compile-verified
//
#include <hip/hip_runtime.h>

typedef __attribute__((ext_vector_type(2))) float v2f;
typedef __attribute__((ext_vector_type(8))) float v8f;
typedef __attribute__((ext_vector_type(4))) unsigned int v4u;
typedef __attribute__((ext_vector_type(8))) int v8i;
typedef __attribute__((ext_vector_type(4))) int v4i;

#define D_DIM 64
#define N_EMB 512
#define T_LEN 8192
#define B_DIM 16
#define WG_THREADS 256
#define WAVES_PER_WG 8
#define Q_PER_WAVE 32          // two 16-row WMMA panels per wave
#define Q_PER_WG (WAVES_PER_WG * Q_PER_WAVE)   // 256 queries per workgroup
#define CB_ROW 528             // padded LDS row stride in floats (512 + 2x8 DW pads)

#if __has_builtin(__builtin_amdgcn_tensor_load_to_lds) && __has_builtin(__builtin_amdgcn_s_wait_tensorcnt)
#define USE_TDM 1
#else
#define USE_TDM 0
#endif

// LDS address of codebook element (d, j) with TDM padding: 8 DWORDs inserted
// after every 256 DWORDs stored -> row stride 528, banks of the two half-waves
// (delta d = 2 -> 1056 DW = 32 banks) are disjoint => conflict-free ds loads.
__device__ __forceinline__ int cbIdx(int d, int j) {
    return d * CB_ROW + j + ((j >> 8) << 3);
}

__global__ __launch_bounds__(WG_THREADS)
void vq_wmma_kernel(const float* __restrict__ zin,
                    const float* __restrict__ cb_all,
                    const long long* __restrict__ cb_idx,
                    float* __restrict__ out)
{
    __shared__ float sCB[D_DIM * CB_ROW];  // ~132 KB padded codebook [d][j]
    __shared__ float sBias[N_EMB];         // -0.5*|c_j|^2
    __shared__ int   sBest[Q_PER_WG];

    const float* cb = cb_all + (size_t)cb_idx[0] * (D_DIM * N_EMB);

    const int tid   = threadIdx.x;
    const int wave  = tid >> 5;
    const int lane  = tid & 31;
    const int half  = lane >> 4;   // 0: lanes 0-15, 1: lanes 16-31
    const int mlane = lane & 15;

#if USE_TDM
    // ---- Tensor Data Mover: DMA 64x512 fp32 codebook -> padded LDS layout ----
    if (wave == 0) {
        const unsigned           ldsa = (unsigned)(unsigned long long)(const void*)sCB;
        const unsigned long long ga   = (unsigned long long)(const void*)cb;
        // D# group0: count=1 | lds_addr | global_addr[56:0] | type=2
        v4u g0 = { 1u,
                   ldsa,
                   (unsigned)(ga & 0xFFFFFFFFull),
                   (unsigned)((ga >> 32) & 0x1FFFFFFull) | 0x80000000u };
        // D# group1: data_size=4B, pad_enable, pad_interval=256DW, pad_amount=8DW;
        // tensor 512x64 (X=j contiguous, Y=d stride 512), tile = whole tensor.
        v8i g1 = { (int)0x0FD20000,          // mask=0|ds=2|pad_en|intv=7|amt=7
                   (int)(512u << 16),        // tensor_dim0 low16 (=512)
                   (int)(64u  << 16),        // tensor_dim1 low16 (=64)
                   (int)(512u << 16),        // tile_dim0 (=512)
                   64,                       // tile_dim1 (=64), tile_dim2=0
                   512,                      // tensor_dim0_stride low32
                   0, 0 };
        v4i gz = { 0, 0, 0, 0 };
#if defined(__clang_major__) && (__clang_major__ >= 23)
        v8i gz8 = { 0, 0, 0, 0, 0, 0, 0, 0 };
        __builtin_amdgcn_tensor_load_to_lds(g0, g1, gz, gz, gz8, 0);
#else
        __builtin_amdgcn_tensor_load_to_lds(g0, g1, gz, gz, 0);
#endif
    }
#else
    // ---- Fallback: manual staging into the same padded layout ----
    {
        const float4* src = (const float4*)cb;
        #pragma unroll 4
        for (int i = tid; i < (D_DIM * N_EMB) / 4; i += WG_THREADS) {
            const int d = i >> 7;            // 128 float4 per row
            const int j = (i & 127) << 2;
            *(float4*)&sCB[cbIdx(d, j)] = src[i];
        }
    }
#endif

    // ---- Column bias from global (L2-resident; overlaps TDM) ----
    for (int j = tid; j < N_EMB; j += WG_THREADS) {
        float s = 0.f;
        #pragma unroll 8
        for (int d = 0; d < D_DIM; ++d) {
            const float c = cb[d * N_EMB + j];
            s += c * c;
        }
        sBias[j] = -0.5f * s;
    }

    const int qbase = blockIdx.x * Q_PER_WG;
    const int b  = qbase / T_LEN;
    const int t0 = (qbase % T_LEN) + wave * Q_PER_WAVE;

    // ---- Two A panels (M=32): 16x4 fp32 fragments, K=64 in 16 chunks ----
    // A frag layout: lanes 0-15 M=0..15 K={0,1}; lanes 16-31 M=0..15 K={2,3}
    const float* zb0 = zin + (size_t)b * D_DIM * T_LEN + (size_t)(t0 + mlane);
    const float* zb1 = zb0 + 16;
    v2f a0[16], a1[16];
    #pragma unroll
    for (int c = 0; c < 16; ++c) {
        const int d0 = 4 * c + 2 * half;
        a0[c].x = zb0[(size_t)d0 * T_LEN];
        a0[c].y = zb0[(size_t)(d0 + 1) * T_LEN];
        a1[c].x = zb1[(size_t)d0 * T_LEN];
        a1[c].y = zb1[(size_t)(d0 + 1) * T_LEN];
    }

#if USE_TDM
    if (wave == 0) __builtin_amdgcn_s_wait_tensorcnt(0);
#endif
    __syncthreads();

    float best0[8], best1[8];
    int   bidx0[8], bidx1[8];
    #pragma unroll
    for (int r = 0; r < 8; ++r) {
        best0[r] = -3.4e38f; bidx0[r] = 0;
        best1[r] = -3.4e38f; bidx1[r] = 0;
    }

    // ---- Sweep 32 N-tiles; each B fragment feeds 2 WMMAs (panels 0 & 1) ----
    #pragma unroll 1
    for (int tile = 0; tile < N_EMB / 16; ++tile) {
        const int n = tile * 16 + mlane;
        const float bias = sBias[n];
        v8f acc0 = {bias, bias, bias, bias, bias, bias, bias, bias};
        v8f acc1 = acc0;
        #pragma unroll
        for (int c = 0; c < 16; ++c) {
            const int d0 = 4 * c + 2 * half;
            // B frag: lanes 0-15 N=0..15 K={0,1}; lanes 16-31 K={2,3}
            v2f bf;
            bf.x = sCB[cbIdx(d0, n)];
            bf.y = sCB[cbIdx(d0 + 1, n)];
            acc0 = __builtin_amdgcn_wmma_f32_16x16x4_f32(
                false, a0[c], false, bf, (short)0, acc0, false, false);
            acc1 = __builtin_amdgcn_wmma_f32_16x16x4_f32(
                false, a1[c], false, bf, (short)0, acc1, false, false);
        }
        #pragma unroll
        for (int r = 0; r < 8; ++r) {
            if (acc0[r] > best0[r]) { best0[r] = acc0[r]; bidx0[r] = n; }
            if (acc1[r] > best1[r]) { best1[r] = acc1[r]; bidx1[r] = n; }
        }
    }

    // ---- Argmax reduction across the 16 lanes of each half-wave ----
    #pragma unroll
    for (int off = 1; off < 16; off <<= 1) {
        #pragma unroll
        for (int r = 0; r < 8; ++r) {
            {
                const float os = __shfl_xor(best0[r], off, 32);
                const int   oi = __shfl_xor(bidx0[r], off, 32);
                if (os > best0[r] || (os == best0[r] && oi < bidx0[r])) {
                    best0[r] = os; bidx0[r] = oi;
                }
            }
            {
                const float os = __shfl_xor(best1[r], off, 32);
                const int   oi = __shfl_xor(bidx1[r], off, 32);
                if (os > best1[r] || (os == best1[r] && oi < bidx1[r])) {
                    best1[r] = os; bidx1[r] = oi;
                }
            }
        }
    }

    float* out_zq  = out;
    float* out_zid = out + (size_t)B_DIM * D_DIM * T_LEN;

    if (lane == 0 || lane == 16) {
        #pragma unroll
        for (int r = 0; r < 8; ++r) {
            const int q0 = half * 8 + r;          // panel 0 query
            const int q1 = 16 + half * 8 + r;     // panel 1 query
            sBest[wave * Q_PER_WAVE + q0] = bidx0[r];
            sBest[wave * Q_PER_WAVE + q1] = bidx1[r];
            out_zid[(size_t)b * T_LEN + (t0 + q0)] = (float)bidx0[r];
            out_zid[(size_t)b * T_LEN + (t0 + q1)] = (float)bidx1[r];
        }
    }
    __syncthreads();

    // ---- Gather winning codes, write z_q in [B, D, T] layout ----
    const int j0 = sBest[wave * Q_PER_WAVE + mlane];
    const int j1 = sBest[wave * Q_PER_WAVE + 16 + mlane];
    float* zq = out_zq + (size_t)b * D_DIM * T_LEN + (size_t)(t0 + mlane);
    #pragma unroll 8
    for (int d = 0; d < D_DIM; d += 2) {
        const int dd = d + half;
        zq[(size_t)dd * T_LEN]      = sCB[cbIdx(dd, j0)];
        zq[(size_t)dd * T_LEN + 16] = sCB[cbIdx(dd, j1)];
    }
}

extern "C" void kernel_launch(void* const* d_in, const int* in_sizes, int n_in,
                              void* d_out, int out_size, void* d_ws, size_t ws_size,
                              hipStream_t stream) {
    const float*     z  = (const float*)d_in[0];
    const float*     cb = (const float*)d_in[1];
    const long long* ci = (const long long*)d_in[2];
    float* out = (float*)d_out;

    const int nblocks = (B_DIM * T_LEN) / Q_PER_WG;   // 512 workgroups
    vq_wmma_kernel<<<nblocks, WG_THREADS, 0, stream>>>(z, cb, ci, out);
}